// TopoEncoder_69561290326837
// MI455X (gfx1250) — compile-verified
//
#include <hip/hip_runtime.h>
#include <stdint.h>

// ---------------------------------------------------------------------------
// MI455X notes:
//  * SpMM is L2-resident gather/scatter (x = 51.2MB << 192MB L2); bound by L2
//    bandwidth + f32 atomic throughput, not FLOPs -> fp32 end to end.
//  * Edge triples streamed through LDS with gfx1250 async-to-LDS copies,
//    double buffered (ASYNCcnt + barrier).
//  * LayerNorm row reductions done on the matrix unit with exact FP32 WMMA
//    (V_WMMA_F32_16X16X4_F32, ones-matrix column-sum trick).
// ---------------------------------------------------------------------------

#define LATDIM 128
#define TILE   1024   // edges staged per LDS buffer

typedef float v2f __attribute__((ext_vector_type(2)));
typedef float v8f __attribute__((ext_vector_type(8)));

// ---- CDNA5 async copy: global -> LDS, 16B per lane, tracked by ASYNCcnt ----
__device__ __forceinline__ void async_copy_b128(unsigned lds_off, const void* gsrc) {
  asm volatile("global_load_async_to_lds_b128 %0, %1, off"
               :: "v"(lds_off), "v"(gsrc) : "memory");
}
__device__ __forceinline__ void wait_async0() {
  asm volatile("s_wait_asynccnt 0" ::: "memory");
}

// ---- direct f32 global atomic adds (no CAS fallback), 4 consecutive floats ----
__device__ __forceinline__ void atomic_add4(float* p, float4 v) {
  asm volatile("global_atomic_add_f32 %0, %1, off\n\t"
               "global_atomic_add_f32 %0, %2, off offset:4\n\t"
               "global_atomic_add_f32 %0, %3, off offset:8\n\t"
               "global_atomic_add_f32 %0, %4, off offset:12"
               :: "v"(p), "v"(v.x), "v"(v.y), "v"(v.z), "v"(v.w) : "memory");
}

// ---------------------------------------------------------------------------
// LayerNorm via FP32 WMMA.
// One wave handles 16 nodes. B(4x16) per K-chunk holds 4 features x 16 nodes
// (column j = node0+j, columns map to lane&15 as in the C/D layout). With
// A = ones(16x4):  D[i][j] = sum_k B[k][j] + C[i][j]  -> per-column feature
// sums, accumulated over 32 chunks (K=128). Column sums are invariant to the
// K-row permutation, so each lane's two loaded features just need to cover
// each (chunk, node) slot exactly once — which they do.
// Also zeroes acc (d_out) and the first scatter target buffer.
// ---------------------------------------------------------------------------
__global__ void __launch_bounds__(256)
ln_wmma_kernel(const float* __restrict__ emb, float* __restrict__ xln,
               float* __restrict__ accz, float* __restrict__ xbz, int nTiles16) {
  const int lane = threadIdx.x & 31;
  const int wv   = (blockIdx.x * blockDim.x + threadIdx.x) >> 5;
  if (wv >= nTiles16) return;

  const int node0 = wv * 16;
  // this lane supplies features (c*4 + koff, c*4 + koff + 1) of node (lane&15)
  const int koff = (lane >> 4) * 2;
  const float* rp = emb + (size_t)(node0 + (lane & 15)) * LATDIM + koff;

  const v2f ones = {1.f, 1.f};
  v8f dsum = {0.f, 0.f, 0.f, 0.f, 0.f, 0.f, 0.f, 0.f};
  v8f dsq  = {0.f, 0.f, 0.f, 0.f, 0.f, 0.f, 0.f, 0.f};

#pragma unroll
  for (int c = 0; c < 32; ++c) {
    v2f b = *(const v2f*)(rp + c * 4);
    dsum = __builtin_amdgcn_wmma_f32_16x16x4_f32(false, ones, false, b,
                                                 (short)0, dsum, false, false);
    v2f b2 = b * b;
    dsq  = __builtin_amdgcn_wmma_f32_16x16x4_f32(false, ones, false, b2,
                                                 (short)0, dsq, false, false);
  }

  // lane l holds column (l&15): full feature sum / sum of squares of its node
  const float s   = dsum[0];
  const float sq  = dsq[0];
  const float mu  = s * (1.f / 128.f);
  const float var = sq * (1.f / 128.f) - mu * mu;
  const float rst = rsqrtf(var + 1e-5f);

  const float4 z4 = {0.f, 0.f, 0.f, 0.f};
  // fully-coalesced normalize: 32 lanes sweep one 512B row per iteration
#pragma unroll 4
  for (int n = 0; n < 16; ++n) {
    const float mun = __shfl(mu, n, 32);
    const float rn  = __shfl(rst, n, 32);
    const size_t off = (size_t)(node0 + n) * LATDIM + lane * 4;
    float4 v = *(const float4*)(emb + off);
    float4 o;
    o.x = (v.x - mun) * rn; o.y = (v.y - mun) * rn;
    o.z = (v.z - mun) * rn; o.w = (v.w - mun) * rn;
    *(float4*)(xln + off)  = o;
    *(float4*)(accz + off) = z4;   // acc = 0
    *(float4*)(xbz + off)  = z4;   // first scatter target = 0
  }
}

// ---------------------------------------------------------------------------
// SpMM: xdst[r] += val * xsrc[c] per edge. Edge triples double-buffered
// through LDS with async-to-LDS copies; gathers hit L2; scatter via f32
// global atomics (4 per lane, 128 floats per edge per wave).
// ---------------------------------------------------------------------------
__global__ void __launch_bounds__(256)
spmm_atomic_kernel(const int* __restrict__ rows, const int* __restrict__ cols,
                   const float* __restrict__ vals, const float* __restrict__ xsrc,
                   float* __restrict__ xdst, int nEdges) {
  __shared__ __align__(16) int   s_row[2][TILE];
  __shared__ __align__(16) int   s_col[2][TILE];
  __shared__ __align__(16) float s_val[2][TILE];

  const int tid  = threadIdx.x;       // 256 threads = 8 waves
  const int lane = tid & 31;
  const int wave = tid >> 5;
  const int nTiles = nEdges / TILE;
  const int G = gridDim.x;

  // prefetch first tile (4 consecutive elements of each array per thread)
  if ((int)blockIdx.x < nTiles) {
    const size_t e0 = (size_t)blockIdx.x * TILE + tid * 4;
    async_copy_b128((unsigned)(uintptr_t)&s_row[0][tid * 4], rows + e0);
    async_copy_b128((unsigned)(uintptr_t)&s_col[0][tid * 4], cols + e0);
    async_copy_b128((unsigned)(uintptr_t)&s_val[0][tid * 4], vals + e0);
  }

  int it = 0;
  for (int t = blockIdx.x; t < nTiles; t += G, ++it) {
    const int cur = it & 1;
    wait_async0();        // my async copies done
    __syncthreads();      // everyone's copies done; prev buffer fully read

    const int nt = t + G; // prefetch next tile into the other half
    if (nt < nTiles) {
      const int nb = cur ^ 1;
      const size_t e0 = (size_t)nt * TILE + tid * 4;
      async_copy_b128((unsigned)(uintptr_t)&s_row[nb][tid * 4], rows + e0);
      async_copy_b128((unsigned)(uintptr_t)&s_col[nb][tid * 4], cols + e0);
      async_copy_b128((unsigned)(uintptr_t)&s_val[nb][tid * 4], vals + e0);
    }

    const int ebase = wave * (TILE / 8);  // 128 edges per wave
#pragma unroll 4
    for (int i = 0; i < TILE / 8; ++i) {
      const int e = ebase + i;
      const int   r = s_row[cur][e];     // LDS broadcast reads
      const int   c = s_col[cur][e];
      const float v = s_val[cur][e];
      const float4 xv = *(const float4*)(xsrc + (size_t)c * LATDIM + lane * 4);
      float4 m;
      m.x = xv.x * v; m.y = xv.y * v; m.z = xv.z * v; m.w = xv.w * v;
      atomic_add4(xdst + (size_t)r * LATDIM + lane * 4, m);
    }
    __syncthreads();      // buffer 'cur' free for reuse two iterations later
  }

  // tail edges (nEdges % TILE), one wave per edge, direct loads
  for (int e = nTiles * TILE + (int)blockIdx.x * 8 + wave; e < nEdges; e += G * 8) {
    const int   r = rows[e];
    const int   c = cols[e];
    const float v = vals[e];
    const float4 xv = *(const float4*)(xsrc + (size_t)c * LATDIM + lane * 4);
    float4 m;
    m.x = xv.x * v; m.y = xv.y * v; m.z = xv.z * v; m.w = xv.w * v;
    atomic_add4(xdst + (size_t)r * LATDIM + lane * 4, m);
  }
}

// ---------------------------------------------------------------------------
// acc += src ; optionally zero the next scatter target in the same sweep.
// ---------------------------------------------------------------------------
__global__ void __launch_bounds__(256)
acc_add_zero_kernel(float* __restrict__ acc, const float* __restrict__ src,
                    float* __restrict__ zbuf, long n4) {
  const long i = (long)blockIdx.x * blockDim.x + threadIdx.x;
  if (i >= n4) return;
  float4 a = ((float4*)acc)[i];
  const float4 s = ((const float4*)src)[i];
  a.x += s.x; a.y += s.y; a.z += s.z; a.w += s.w;
  ((float4*)acc)[i] = a;
  if (zbuf) {
    const float4 z = {0.f, 0.f, 0.f, 0.f};
    ((float4*)zbuf)[i] = z;
  }
}

extern "C" void kernel_launch(void* const* d_in, const int* in_sizes, int n_in,
                              void* d_out, int out_size, void* d_ws, size_t ws_size,
                              hipStream_t stream) {
  (void)n_in; (void)out_size; (void)ws_size;
  const float* emb  = (const float*)d_in[0];
  const int*   arow = (const int*)d_in[1];
  const int*   acol = (const int*)d_in[2];
  const float* aval = (const float*)d_in[3];

  const int N = in_sizes[0] / LATDIM;   // 100000
  const int E = in_sizes[1];            // 640000

  float* acc = (float*)d_out;
  float* xA  = (float*)d_ws;                       // ping
  float* xB  = xA + (size_t)N * LATDIM;            // pong

  const int  tiles16  = N / 16;                    // 6250 (N % 16 == 0)
  const int  lnBlocks = (tiles16 * 32 + 255) / 256;
  const long n4       = (long)N * LATDIM / 4;
  const int  addBlks  = (int)((n4 + 255) / 256);
  const int  spmmBlks = 320;

  // x0 = LN(embeds) -> xA ; acc = 0 ; xB = 0
  ln_wmma_kernel<<<lnBlocks, 256, 0, stream>>>(emb, xA, acc, xB, tiles16);

  // hop 1: xB = A @ xA ; acc += xB ; xA = 0
  spmm_atomic_kernel<<<spmmBlks, 256, 0, stream>>>(arow, acol, aval, xA, xB, E);
  acc_add_zero_kernel<<<addBlks, 256, 0, stream>>>(acc, xB, xA, n4);

  // hop 2: xA = A @ xB ; acc += xA ; xB = 0
  spmm_atomic_kernel<<<spmmBlks, 256, 0, stream>>>(arow, acol, aval, xB, xA, E);
  acc_add_zero_kernel<<<addBlks, 256, 0, stream>>>(acc, xA, xB, n4);

  // hop 3: xB = A @ xA ; acc += xB
  spmm_atomic_kernel<<<spmmBlks, 256, 0, stream>>>(arow, acol, aval, xA, xB, E);
  acc_add_zero_kernel<<<addBlks, 256, 0, stream>>>(acc, xB, nullptr, n4);
}